// Sampler_39848706573590
// MI455X (gfx1250) — compile-verified
//
#include <hip/hip_runtime.h>
#include <stdint.h>

#define THREADS 512
#define NW 16            // 512 / wave32
#define NB1 2048         // 11-bit digit histograms (levels 1,2)
#define NB3 1024         // 10-bit digit histogram (level 3)

// ---------- order-preserving float <-> uint key ----------
__device__ __forceinline__ uint32_t f2key(float x) {
  uint32_t u = __float_as_uint(x);
  return (u & 0x80000000u) ? ~u : (u | 0x80000000u);
}
__device__ __forceinline__ float key2f(uint32_t k) {
  uint32_t u = (k & 0x80000000u) ? (k & 0x7fffffffu) : ~k;
  return __uint_as_float(u);
}

// ---------- wave32 primitives ----------
__device__ __forceinline__ float wsumf(float v) {
  #pragma unroll
  for (int o = 16; o; o >>= 1) v += __shfl_xor(v, o, 32);
  return v;
}
__device__ __forceinline__ uint32_t wsumu(uint32_t v) {
  #pragma unroll
  for (int o = 16; o; o >>= 1) v += (uint32_t)__shfl_xor((int)v, o, 32);
  return v;
}
__device__ __forceinline__ void wargmax(float& v, int& i) {
  #pragma unroll
  for (int o = 16; o; o >>= 1) {
    float ov = __shfl_xor(v, o, 32);
    int   oi = __shfl_xor(i, o, 32);
    if (ov > v || (ov == v && oi < i)) { v = ov; i = oi; }
  }
}

// ---------- block primitives (barriers reached uniformly) ----------
__device__ __forceinline__ float block_sumf(float v, float* sh, int tid) {
  v = wsumf(v);
  int wid = tid >> 5, lane = tid & 31;
  if (lane == 0) sh[wid] = v;
  __syncthreads();
  if (tid < 32) {
    float s = (tid < NW) ? sh[tid] : 0.f;
    s = wsumf(s);
    if (tid == 0) sh[0] = s;
  }
  __syncthreads();
  v = sh[0];
  __syncthreads();
  return v;
}
__device__ __forceinline__ uint32_t block_sumu(uint32_t v, int* sh, int tid) {
  v = wsumu(v);
  int wid = tid >> 5, lane = tid & 31;
  if (lane == 0) sh[wid] = (int)v;
  __syncthreads();
  if (tid < 32) {
    uint32_t s = (tid < NW) ? (uint32_t)sh[tid] : 0u;
    s = wsumu(s);
    if (tid == 0) sh[0] = (int)s;
  }
  __syncthreads();
  v = (uint32_t)sh[0];
  __syncthreads();
  return v;
}
__device__ __forceinline__ void block_argmax(float& v, int& i, float* shv, int* shi, int tid) {
  wargmax(v, i);
  int wid = tid >> 5, lane = tid & 31;
  if (lane == 0) { shv[wid] = v; shi[wid] = i; }
  __syncthreads();
  if (tid < 32) {
    float vv = (tid < NW) ? shv[tid] : -__builtin_inff();
    int   ii = (tid < NW) ? shi[tid] : 0x7fffffff;
    wargmax(vv, ii);
    if (tid == 0) { shv[0] = vv; shi[0] = ii; }
  }
  __syncthreads();
  v = shv[0]; i = shi[0];
  __syncthreads();
}

// ---------- suffix (descending) select over a histogram ----------
// Finds bin b with  countStrictlyAbove(b) < target <= countStrictlyAbove(b)+hist[b]
__device__ __forceinline__ void sel_count(const uint32_t* hist, int nbins, uint32_t target,
                                          uint32_t* gsum, int* obin, uint32_t* orem, int tid) {
  int ng = nbins >> 5, wid = tid >> 5, lane = tid & 31;
  for (int g = wid; g < ng; g += NW) {
    uint32_t c = hist[(g << 5) + lane];
    c = wsumu(c);
    if (lane == 0) gsum[g] = c;
  }
  __syncthreads();
  if (tid == 0) {
    uint32_t cum = 0; int G = 0;
    for (int g = ng - 1; g >= 0; --g) {
      uint32_t c = gsum[g];
      if (cum + c >= target) { G = g; break; }
      cum += c;
    }
    int b = G << 5;
    for (int j = 31; j >= 0; --j) {
      uint32_t c = hist[(G << 5) + j];
      if (cum + c >= target) { b = (G << 5) + j; break; }
      cum += c;
    }
    *obin = b;
    *orem = (target > cum) ? (target - cum) : 1u;
  }
  __syncthreads();
}
__device__ __forceinline__ void sel_w(const float* hist, int nbins, float target,
                                      float* gsum, int* obin, float* orem, int tid) {
  int ng = nbins >> 5, wid = tid >> 5, lane = tid & 31;
  for (int g = wid; g < ng; g += NW) {
    float c = hist[(g << 5) + lane];
    c = wsumf(c);
    if (lane == 0) gsum[g] = c;
  }
  __syncthreads();
  if (tid == 0) {
    float cum = 0.f; int G = 0;
    for (int g = ng - 1; g >= 0; --g) {
      float c = gsum[g];
      if (cum + c >= target) { G = g; break; }
      cum += c;
    }
    int b = G << 5;
    for (int j = 31; j >= 0; --j) {
      float c = hist[(G << 5) + j];
      if (cum + c >= target) { b = (G << 5) + j; break; }
      cum += c;
    }
    *obin = b;
    *orem = fmaxf(target - cum, 1e-35f);
  }
  __syncthreads();
}

// Streaming loop over one row (scalar tail for generality)
#define ROW_FOREACH(XV, GIV, ...)                                              \
  { const int n4_ = V >> 2; const float4* L4_ = (const float4*)rowL;           \
    for (int i_ = tid; i_ < n4_; i_ += THREADS) {                              \
      float4 v4_ = L4_[i_];                                                    \
      _Pragma("unroll") for (int c_ = 0; c_ < 4; ++c_) {                       \
        float XV = (c_==0)?v4_.x:(c_==1)?v4_.y:(c_==2)?v4_.z:v4_.w;            \
        int GIV = (i_ << 2) + c_; (void)GIV; __VA_ARGS__ } }                   \
    for (int i_ = (n4_ << 2) + tid; i_ < V; i_ += THREADS) {                   \
      float XV = rowL[i_]; int GIV = i_; (void)GIV; __VA_ARGS__ } }

extern "C" __global__ void __launch_bounds__(THREADS)
sampler_gfx1250(const float* __restrict__ logits,
                const float* __restrict__ temp_p,
                const float* __restrict__ minp_p,
                const int*   __restrict__ topk_p,
                const float* __restrict__ topp_p,
                const float* __restrict__ q,
                const int*   __restrict__ nlp_p,
                float* __restrict__ out,
                int B, int V) {
  __shared__ __align__(16) float stage[2][THREADS * 4];   // async-DMA double buffer
  __shared__ uint32_t histA[NB1];
  __shared__ uint32_t histB[NB1];
  __shared__ uint32_t histC[NB1];   // pass-1 privatized copies (contention /4)
  __shared__ uint32_t histD[NB1];
  __shared__ uint32_t gsum_u[NB1 / 32];
  __shared__ float    gsum_f[NB1 / 32];
  __shared__ float    shf[NW];
  __shared__ int      shi[NW];
  __shared__ int      s_bin;
  __shared__ uint32_t s_rem;
  __shared__ float    s_remf;
  __shared__ float    cval[64];
  __shared__ int      cidx[64];
  __shared__ int      ccnt;
  __shared__ float    o_lp[65];
  __shared__ int      o_idx[65];
  __shared__ float    Sm, SlogZ, Sthr2, SthrF, St20, Sxs;
  __shared__ int      Sgreedy, Ssampled;
  __shared__ uint32_t Sb1k, Sb1t, Sb2k, Sb2t, Sremk, Sremt, Sb1p, Sb2p;
  __shared__ float    Sremp;

  const int tid = threadIdx.x;
  const int row = blockIdx.x;
  const float* rowL = logits + (size_t)row * V;
  const float* rowQ = q + (size_t)row * V;
  const float NEG = -__builtin_inff();
  float* histF = (float*)histA;   // reuse as float histogram

  // ================= Pass 1: HBM -> LDS async DMA stream =================
  // max m, greedy argmax (first index), level-1 count histogram (11 MSB digits)
  for (int i = tid; i < NB1; i += THREADS) {
    histA[i] = 0u; histB[i] = 0u; histC[i] = 0u; histD[i] = 0u;
  }
  __syncthreads();

  const int tileF = THREADS * 4;
  const int ntiles = (V + tileF - 1) / tileF;
  const uint32_t lds0 = (uint32_t)(size_t)&stage[0][0];
  const uint32_t lds1 = (uint32_t)(size_t)&stage[1][0];
  const long maxByte = (long)V * 4 - 16;

  auto issue = [&](const float* base_ptr, int t, int buf) {
    long b = ((long)t * tileF + tid * 4) * 4;
    if (b > maxByte) b = maxByte;
    if (b < 0) b = 0;
    uint32_t lds = (buf ? lds1 : lds0) + (uint32_t)(tid * 16);
    uint32_t off = (uint32_t)b;
    // GVS mode: saddr (row base) + 32-bit per-lane byte offset -> per-lane LDS addr
    asm volatile("global_load_async_to_lds_b128 %0, %1, %2"
                 :: "v"(lds), "v"(off), "s"(base_ptr) : "memory");
  };

  // per-wave-group private histogram copy
  const int hsel = (tid >> 5) & 3;
  uint32_t* const myh = (hsel == 0) ? histA : (hsel == 1) ? histB
                       : (hsel == 2) ? histC : histD;

  float lmax = NEG; int lidx = 0x7fffffff;
  issue(rowL, 0, 0);
  for (int t = 0; t < ntiles; ++t) {
    if (t + 1 < ntiles) {
      issue(rowL, t + 1, (t + 1) & 1);                  // prefetch next tile
      asm volatile("s_wait_asynccnt 0x1" ::: "memory"); // current tile landed
    } else {
      asm volatile("s_wait_asynccnt 0x0" ::: "memory");
    }
    const float4 x4 = *(const float4*)&stage[t & 1][tid * 4];
    long base = (long)t * tileF + tid * 4;
    #pragma unroll
    for (int c = 0; c < 4; ++c) {
      long gi = base + c;
      if (gi < V) {
        float x = (c==0)?x4.x:(c==1)?x4.y:(c==2)?x4.z:x4.w;
        if (x > lmax) { lmax = x; lidx = (int)gi; }
        atomicAdd(&myh[f2key(x) >> 21], 1u);
      }
    }
  }
  __syncthreads();
  // merge the 4 private copies into histA
  for (int i = tid; i < NB1; i += THREADS)
    histA[i] = histA[i] + histB[i] + histC[i] + histD[i];
  __syncthreads();

  block_argmax(lmax, lidx, shf, shi, tid);
  if (tid == 0) { Sm = lmax; Sgreedy = lidx; }
  __syncthreads();

  const float m   = Sm;
  const float T0  = temp_p[row];
  const float Tc  = fmaxf(T0, 1e-6f);
  const float invT = 1.0f / Tc;
  int k = topk_p[row]; k = k < 1 ? 1 : (k > V ? V : k);
  int nlp = *nlp_p;    nlp = nlp < 1 ? 1 : (nlp > 64 ? 64 : nlp);
  const float thr1 = m + Tc * __logf(fmaxf(minp_p[row], 1e-38f));  // min-p threshold in logit space

  // level-1 digit for k-th largest and nlp-th largest (shared histogram)
  sel_count(histA, NB1, (uint32_t)k, gsum_u, &s_bin, &s_rem, tid);
  if (tid == 0) { Sb1k = (uint32_t)s_bin; Sremk = s_rem; }
  __syncthreads();
  sel_count(histA, NB1, (uint32_t)nlp, gsum_u, &s_bin, &s_rem, tid);
  if (tid == 0) { Sb1t = (uint32_t)s_bin; Sremt = s_rem; }
  __syncthreads();

  // ================= Pass 2 (L2-resident): Zraw + level-2 digit histograms ====
  for (int i = tid; i < NB1; i += THREADS) { histA[i] = 0u; histB[i] = 0u; }
  __syncthreads();
  {
    const uint32_t b1k = Sb1k, b1t = Sb1t;
    float zs = 0.f;
    ROW_FOREACH(x, gi, {
      zs += __expf(x - m);
      uint32_t key = f2key(x);
      uint32_t hi = key >> 21;
      if (hi == b1k) atomicAdd(&histA[(key >> 10) & 0x7FFu], 1u);
      if (hi == b1t) atomicAdd(&histB[(key >> 10) & 0x7FFu], 1u);
    })
    __syncthreads();
    zs = block_sumf(zs, shf, tid);
    if (tid == 0) SlogZ = __logf(zs);
  }
  sel_count(histA, NB1, Sremk, gsum_u, &s_bin, &s_rem, tid);
  if (tid == 0) { Sb2k = (uint32_t)s_bin; Sremk = s_rem; }
  __syncthreads();
  sel_count(histB, NB1, Sremt, gsum_u, &s_bin, &s_rem, tid);
  if (tid == 0) { Sb2t = (uint32_t)s_bin; Sremt = s_rem; }
  __syncthreads();

  // ================= Pass 3: level-3 digits -> exact kth / t20 values =========
  for (int i = tid; i < NB1; i += THREADS) { histA[i] = 0u; histB[i] = 0u; }
  __syncthreads();
  {
    const uint32_t pk = (Sb1k << 11) | Sb2k;
    const uint32_t pt = (Sb1t << 11) | Sb2t;
    ROW_FOREACH(x, gi, {
      uint32_t key = f2key(x);
      uint32_t hi22 = key >> 10;
      if (hi22 == pk) atomicAdd(&histA[key & 0x3FFu], 1u);
      if (hi22 == pt) atomicAdd(&histB[key & 0x3FFu], 1u);
    })
    __syncthreads();
    sel_count(histA, NB3, Sremk, gsum_u, &s_bin, &s_rem, tid);
    if (tid == 0) {
      float kv = key2f((pk << 10) | (uint32_t)s_bin);
      // top-k only bites if the k-th overall survives min-p; combined => max()
      Sthr2 = fmaxf(thr1, kv);
    }
    __syncthreads();
    sel_count(histB, NB3, Sremt, gsum_u, &s_bin, &s_rem, tid);
    if (tid == 0) St20 = key2f((pt << 10) | (uint32_t)s_bin);
    __syncthreads();
  }

  // ================= Pass 4: filtered softmax mass + weighted level-1 =========
  // Survivors of thr2 number at most ~top_k (<=1024): atomics here are cheap.
  for (int i = tid; i < NB1; i += THREADS) histF[i] = 0.f;
  __syncthreads();
  const float thr2 = Sthr2;
  {
    float zf = 0.f;
    ROW_FOREACH(x, gi, {
      if (x >= thr2) {
        float w = __expf((x - m) * invT);
        zf += w;
        atomicAdd(&histF[f2key(x) >> 21], w);
      }
    })
    __syncthreads();
    zf = block_sumf(zf, shf, tid);
    float targ = fmaxf(topp_p[row] * zf, 1e-35f);  // keep-max guaranteed
    sel_w(histF, NB1, targ, gsum_f, &s_bin, &s_remf, tid);
    if (tid == 0) { Sb1p = (uint32_t)s_bin; Sremp = s_remf; }
    __syncthreads();
  }
  // ================= Pass 5: weighted level-2 =================
  for (int i = tid; i < NB1; i += THREADS) histF[i] = 0.f;
  __syncthreads();
  {
    const uint32_t b1p = Sb1p;
    ROW_FOREACH(x, gi, {
      if (x >= thr2) {
        uint32_t key = f2key(x);
        if ((key >> 21) == b1p) atomicAdd(&histF[(key >> 10) & 0x7FFu], __expf((x - m) * invT));
      }
    })
    __syncthreads();
    sel_w(histF, NB1, Sremp, gsum_f, &s_bin, &s_remf, tid);
    if (tid == 0) { Sb2p = (uint32_t)s_bin; Sremp = s_remf; }
    __syncthreads();
  }
  // ================= Pass 6: weighted level-3 -> top-p threshold ==============
  for (int i = tid; i < NB3; i += THREADS) histF[i] = 0.f;
  __syncthreads();
  {
    const uint32_t pp = (Sb1p << 11) | Sb2p;
    ROW_FOREACH(x, gi, {
      if (x >= thr2) {
        uint32_t key = f2key(x);
        if ((key >> 10) == pp) atomicAdd(&histF[key & 0x3FFu], __expf((x - m) * invT));
      }
    })
    __syncthreads();
    sel_w(histF, NB3, Sremp, gsum_f, &s_bin, &s_remf, tid);
    if (tid == 0) {
      SthrF = fmaxf(thr2, key2f((pp << 10) | (uint32_t)s_bin));
      ccnt = 0;
    }
    __syncthreads();
  }

  // ================= Pass 7: Gumbel-race argmax + top-nlp candidates ==========
  // q is the second first-touch HBM stream: feed it through the async DMA
  // double-buffer (stage[] is free again); logits come from L2 directly.
  const float thrF = SthrF, t20v = St20;
  float best = NEG; int bi = 0x7fffffff;
  issue(rowQ, 0, 0);
  for (int t = 0; t < ntiles; ++t) {
    if (t + 1 < ntiles) {
      issue(rowQ, t + 1, (t + 1) & 1);
      asm volatile("s_wait_asynccnt 0x1" ::: "memory");
    } else {
      asm volatile("s_wait_asynccnt 0x0" ::: "memory");
    }
    const float4 q4 = *(const float4*)&stage[t & 1][tid * 4];
    long base = (long)t * tileF + tid * 4;
    if (base + 4 <= (long)V) {
      const float4 x4 = *(const float4*)(rowL + base);
      #pragma unroll
      for (int c = 0; c < 4; ++c) {
        float x  = (c==0)?x4.x:(c==1)?x4.y:(c==2)?x4.z:x4.w;
        float qq = (c==0)?q4.x:(c==1)?q4.y:(c==2)?q4.z:q4.w;
        int gi = (int)base + c;
        if (x >= thrF) {
          float e = -__logf(fminf(fmaxf(qq, 1e-10f), 1.0f));
          float sc = x * invT - __logf(e);
          if (sc > best) { best = sc; bi = gi; }
        }
        if (x >= t20v) {
          int p = atomicAdd(&ccnt, 1);
          if (p < 64) { cval[p] = x; cidx[p] = gi; }
        }
      }
    } else {
      #pragma unroll
      for (int c = 0; c < 4; ++c) {
        long gi = base + c;
        if (gi < V) {
          float x  = rowL[gi];
          float qq = rowQ[gi];   // safe scalar path for a ragged tail
          if (x >= thrF) {
            float e = -__logf(fminf(fmaxf(qq, 1e-10f), 1.0f));
            float sc = x * invT - __logf(e);
            if (sc > best) { best = sc; bi = (int)gi; }
          }
          if (x >= t20v) {
            int p = atomicAdd(&ccnt, 1);
            if (p < 64) { cval[p] = x; cidx[p] = (int)gi; }
          }
        }
      }
    }
  }
  block_argmax(best, bi, shf, shi, tid);
  if (tid == 0) {
    int samp = (T0 < 1e-5f) ? Sgreedy : bi;
    Ssampled = samp;
    Sxs = rowL[samp];
  }
  __syncthreads();

  // ================= Pass 8: rank of sampled token ============================
  uint32_t cr = 0;
  {
    const float xs = Sxs;
    ROW_FOREACH(x, gi, { if (x >= xs) cr++; })
  }
  cr = block_sumu(cr, shi, tid);
  const uint32_t rank = cr;

  // ================= Epilogue: sort candidates, emit outputs ==================
  int cnt = ccnt; cnt = cnt > 64 ? 64 : cnt;
  for (int c = tid; c < nlp + 1; c += THREADS) { o_idx[c] = 0; o_lp[c] = NEG; }
  __syncthreads();
  const float logZ = SlogZ;
  if (tid < cnt) {
    float v = cval[tid]; int id = cidx[tid];
    int r = 0;
    for (int j = 0; j < cnt; ++j) {
      float vj = cval[j]; int ij = cidx[j];
      if (vj > v || (vj == v && ij < id)) r++;
    }
    if (r < nlp) { o_idx[1 + r] = id; o_lp[1 + r] = v - m - logZ; }
  }
  if (tid == 0) { o_idx[0] = Ssampled; o_lp[0] = Sxs - m - logZ; }
  __syncthreads();

  const int W = nlp + 1;
  float* out_s = out;                               // (B,1) sampled
  float* out_i = out + B;                           // (B,W) indices
  float* out_l = out + B + (size_t)B * W;           // (B,W) logprobs
  float* out_r = out + B + (size_t)2 * B * W;       // (B,)  ranks
  if (tid == 0) { out_s[row] = (float)Ssampled; out_r[row] = (float)rank; }
  for (int c = tid; c < W; c += THREADS) {
    out_i[(size_t)row * W + c] = (float)o_idx[c];
    out_l[(size_t)row * W + c] = o_lp[c];
  }
}

extern "C" void kernel_launch(void* const* d_in, const int* in_sizes, int n_in,
                              void* d_out, int out_size, void* d_ws, size_t ws_size,
                              hipStream_t stream) {
  (void)n_in; (void)out_size; (void)d_ws; (void)ws_size;
  const float* logits      = (const float*)d_in[0];
  const float* temperature = (const float*)d_in[1];
  const float* min_p       = (const float*)d_in[2];
  const int*   top_k       = (const int*)d_in[3];
  const float* top_p       = (const float*)d_in[4];
  const float* q           = (const float*)d_in[5];
  const int*   num_lp      = (const int*)d_in[6];
  int B = in_sizes[1];
  int V = (B > 0) ? (in_sizes[0] / B) : 0;
  if (B <= 0 || V <= 0) return;
  sampler_gfx1250<<<B, THREADS, 0, stream>>>(logits, temperature, min_p, top_k,
                                             top_p, q, num_lp, (float*)d_out, B, V);
}